// LineEvo_33603824124404
// MI455X (gfx1250) — compile-verified
//
#include <hip/hip_runtime.h>
#include <hip/hip_bf16.h>
#include <cmath>

typedef __attribute__((ext_vector_type(2))) float v2f;
typedef __attribute__((ext_vector_type(8))) float v8f;

#define C_IN   63
#define C_D    128
#define C_NG   6

// ---------------------------------------------------------------------------
// Kernel 0: initialize output. out layout: G x 256  ([0,128)=sum, [128,256)=max)
// ---------------------------------------------------------------------------
__global__ void init_out_kernel(float* __restrict__ out, int n) {
    int i = blockIdx.x * blockDim.x + threadIdx.x;
    if (i < n) out[i] = ((i & 255) < 128) ? 0.0f : -__builtin_inff();
}

// ---------------------------------------------------------------------------
// Kernel 1: h = x @ W + b via V_WMMA_F32_16X16X4_F32 (full fp32 precision).
// One block = one 16-row M-tile; 8 waves cover the 8 N-tiles of D=128.
// x tile (16 x 63, zero-padded to 64) staged in LDS and shared by all waves.
// ---------------------------------------------------------------------------
__global__ void __launch_bounds__(256)
gemm_wmma_kernel(const float* __restrict__ x, const float* __restrict__ W,
                 const float* __restrict__ bias, float* __restrict__ h, int nrows) {
    __shared__ float xs[16][64];

    const int tid = threadIdx.x;
    const int m0  = blockIdx.x * 16;

    // Stage the 16x63 x-tile (padded to K=64 with zeros) into LDS.
    for (int i = tid; i < 16 * 64; i += 256) {
        int r = i >> 6, c = i & 63;
        int row = m0 + r;
        float v = 0.0f;
        if (c < C_IN && row < nrows) v = x[row * C_IN + c];
        xs[r][c] = v;
    }
    __syncthreads();

    const int lane  = tid & 31;
    const int wv    = tid >> 5;              // 8 waves -> 8 N-tiles (D=128)
    const int n0    = wv * 16;
    const int col   = n0 + (lane & 15);
    const int khalf = (lane >> 4) << 1;      // lanes 0-15: +0 ; lanes 16-31: +2
    const int mrow  = lane & 15;

    v8f acc = {};
    #pragma unroll
    for (int k = 0; k < 64; k += 4) {
        const int kb = k + khalf;
        v2f a, bm;
        // A 16x4 f32: VGPR0 = K=kb, VGPR1 = K=kb+1 (per-lane rows, split halves)
        a.x = xs[mrow][kb];
        a.y = xs[mrow][kb + 1];
        // B 4x16 f32: row-striped across lanes; VGPR0 = rows kb, VGPR1 = rows kb+1
        bm.x = (kb     < C_IN) ? W[kb * C_D + col]       : 0.0f;
        bm.y = (kb + 1 < C_IN) ? W[(kb + 1) * C_D + col] : 0.0f;
        acc = __builtin_amdgcn_wmma_f32_16x16x4_f32(false, a, false, bm,
                                                    (short)0, acc, false, false);
    }

    const float bc    = bias[col];
    const int   rbase = m0 + ((lane >> 4) << 3);   // upper lanes hold M = v+8
    #pragma unroll
    for (int v = 0; v < 8; ++v) {
        int row = rbase + v;
        if (row < nrows) h[row * C_D + col] = acc[v] + bc;
    }
}

// ---------------------------------------------------------------------------
// Kernel 2: per-edge pipeline. One wave (32 lanes) per edge; each lane owns
// 4 of the 128 channels (float4). Score dot-product via wave shuffle reduce.
// Segment sum via atomicAdd, segment max via signed/unsigned atomic bit-trick.
// ---------------------------------------------------------------------------
__device__ __forceinline__ float elu1(float v) {
    return v > 0.0f ? v : (__expf(v) - 1.0f);
}

__device__ __forceinline__ void atomic_max_f32(float* addr, float v) {
    if (v >= 0.0f) atomicMax((int*)addr, __float_as_int(v));
    else           atomicMin((unsigned int*)addr, __float_as_uint(v));
}

__global__ void __launch_bounds__(256)
edge_kernel(const float* __restrict__ h, const float* __restrict__ pos,
            const float* __restrict__ attn, const float* __restrict__ W_rbf,
            const float* __restrict__ W_read, const float* __restrict__ b_read,
            const long long* __restrict__ edges, const long long* __restrict__ batch,
            float* __restrict__ out, int nedges) {
    __shared__ float s_attn[C_D];
    __shared__ float s_wrbf[C_NG * C_D];
    __shared__ float s_wread[C_D];

    const int tid = threadIdx.x;
    for (int i = tid; i < C_D; i += blockDim.x) {
        s_attn[i]  = attn[i];
        s_wread[i] = W_read[i];
    }
    for (int i = tid; i < C_NG * C_D; i += blockDim.x) s_wrbf[i] = W_rbf[i];
    __syncthreads();

    const float brd    = b_read[0];
    const int   lane   = tid & 31;
    const int   wave   = blockIdx.x * (blockDim.x >> 5) + (tid >> 5);
    const int   nwaves = gridDim.x * (blockDim.x >> 5);
    const int   c0     = lane << 2;                // this lane's 4 channels

    for (int e = wave; e < nedges; e += nwaves) {
        const long long src = edges[2 * (long long)e + 0];
        const long long dst = edges[2 * (long long)e + 1];

        // Edge geometry: dist = max(||pos[dst]-pos[src]||, 0.1)
        const float dx = pos[dst * 3 + 0] - pos[src * 3 + 0];
        const float dy = pos[dst * 3 + 1] - pos[src * 3 + 1];
        const float dz = pos[dst * 3 + 2] - pos[src * 3 + 2];
        const float dist = fmaxf(__builtin_sqrtf(dx * dx + dy * dy + dz * dz), 0.1f);

        // RBF: offsets = 0..5 step 1, coeff = -0.5
        float rbf[C_NG];
        #pragma unroll
        for (int g = 0; g < C_NG; ++g) {
            float t = dist - (float)g;
            rbf[g] = __expf(-0.5f * t * t);
        }

        // Gather h[src], h[dst] (L2-resident), message + RBF envelope + attn ELU
        const float4 hs = *(const float4*)(h + src * C_D + c0);
        const float4 hd = *(const float4*)(h + dst * C_D + c0);
        float he[4] = { elu1(hs.x + hd.x), elu1(hs.y + hd.y),
                        elu1(hs.z + hd.z), elu1(hs.w + hd.w) };

        float a4[4];
        #pragma unroll
        for (int j = 0; j < 4; ++j) {
            float w = 0.0f;
            #pragma unroll
            for (int g = 0; g < C_NG; ++g) w += rbf[g] * s_wrbf[g * C_D + c0 + j];
            a4[j] = elu1(he[j] * w * s_attn[c0 + j]);
        }

        // score = sigmoid(atom . W_read + b_read): wave32 shuffle reduction
        float part = a4[0] * s_wread[c0 + 0] + a4[1] * s_wread[c0 + 1] +
                     a4[2] * s_wread[c0 + 2] + a4[3] * s_wread[c0 + 3];
        #pragma unroll
        for (int off = 16; off > 0; off >>= 1) part += __shfl_xor(part, off, 32);
        const float score = 1.0f / (1.0f + __expf(-(part + brd)));

        // Segment reduce into out[g][0:128]=sum, out[g][128:256]=max
        const long long seg  = batch[src];
        float*          base = out + seg * 256;
        #pragma unroll
        for (int j = 0; j < 4; ++j) {
            atomicAdd(base + c0 + j, a4[j] * score);
            atomic_max_f32(base + 128 + c0 + j, a4[j]);
        }
    }
}

// ---------------------------------------------------------------------------
// Kernel 3: replace never-touched segment-max slots (-inf) with 0.
// ---------------------------------------------------------------------------
__global__ void fix_max_kernel(float* __restrict__ out, int n) {   // n = G*128
    int i = blockIdx.x * blockDim.x + threadIdx.x;
    if (i < n) {
        float* p = out + (long long)(i >> 7) * 256 + 128 + (i & 127);
        float v = *p;
        if (!__builtin_isfinite(v)) *p = 0.0f;
    }
}

// ---------------------------------------------------------------------------
extern "C" void kernel_launch(void* const* d_in, const int* in_sizes, int n_in,
                              void* d_out, int out_size, void* d_ws, size_t ws_size,
                              hipStream_t stream) {
    const float*     x      = (const float*)d_in[0];
    const float*     pos    = (const float*)d_in[1];
    const float*     W      = (const float*)d_in[2];
    const float*     b      = (const float*)d_in[3];
    const float*     attn   = (const float*)d_in[4];
    const float*     W_rbf  = (const float*)d_in[5];
    const float*     W_read = (const float*)d_in[6];
    const float*     b_read = (const float*)d_in[7];
    const long long* edges  = (const long long*)d_in[8];
    const long long* batch  = (const long long*)d_in[9];
    float*           out    = (float*)d_out;
    float*           h      = (float*)d_ws;      // N x 128 f32 (~102 MB)

    const int nnodes = in_sizes[0] / C_IN;       // 200000
    const int nedges = in_sizes[8] / 2;          // 400000

    // 0) init output halves (sum=0, max=-inf)
    init_out_kernel<<<(out_size + 255) / 256, 256, 0, stream>>>(out, out_size);

    // 1) h = x @ W + b  (fp32 WMMA)
    const int mtiles = (nnodes + 15) / 16;
    gemm_wmma_kernel<<<mtiles, 256, 0, stream>>>(x, W, b, h, nnodes);

    // 2) per-edge message/attention/readout + segment atomics
    edge_kernel<<<2048, 256, 0, stream>>>(h, pos, attn, W_rbf, W_read, b_read,
                                          edges, batch, out, nedges);

    // 3) -inf -> 0 in the max half
    const int nmax = out_size / 2;
    fix_max_kernel<<<(nmax + 255) / 256, 256, 0, stream>>>(out, nmax);
}